// SGNN_2482491097660
// MI455X (gfx1250) — compile-verified
//
#include <hip/hip_runtime.h>
#include <hip/hip_bf16.h>

// ---------------------------------------------------------------------------
// SGNN pipeline for MI455X (gfx1250, wave32, WMMA + TDM).
// Edge layers:  ea' = relu([ (h[src]+h[dst])@Wx+bx , ea@We+be ])     rows = E
// Node layers:  h'  = relu([ h@Wx+bx , segsum(ea,src)@We+be ])       rows = N
// Final:        out = [ broadcast(sum_n h) , h ]                     [N,12]
//
// GEMMs: V_WMMA_F32_16X16X4_F32 (fp32-exact), dims compile-time, weights
// zero-padded + pair-interleaved in LDS (B frag = one ds_load_b64), loops
// fully unrolled. Second-GEMM operand tile prefetched at kernel entry via
// TENSOR_LOAD_TO_LDS (iterate descriptor, 1 row x C2, 16 iterations,
// lds stride ASTR) overlapping gather+GEMM1; s_wait_tensorcnt before GEMM2.
// Output tile staged in LDS, flushed as one contiguous float4 stream.
// ---------------------------------------------------------------------------

typedef float v2f __attribute__((ext_vector_type(2)));
typedef float v8f __attribute__((ext_vector_type(8)));
typedef unsigned int u32x4 __attribute__((ext_vector_type(4)));
typedef int i32x4 __attribute__((ext_vector_type(4)));
typedef int i32x8 __attribute__((ext_vector_type(8)));

#define WAVES_PER_BLOCK 8
#define TILE_M 16
#define ROWS_PER_BLOCK (WAVES_PER_BLOCK * TILE_M)
#define ASTR 68   // A-tile LDS row stride (floats): gcd(68,64)=4 -> conflict-free

#if defined(__HIP_DEVICE_COMPILE__) &&                        \
    __has_builtin(__builtin_amdgcn_tensor_load_to_lds) &&     \
    __has_builtin(__builtin_amdgcn_s_wait_tensorcnt)
#define HAVE_TDM 1
#else
#define HAVE_TDM 0
#endif

#if __has_include(<hip/amd_detail/amd_gfx1250_TDM.h>)
#define TDM_ARITY6 1   // amdgpu-toolchain / therock headers -> 6-arg builtin
#else
#define TDM_ARITY6 0   // ROCm 7.2 -> 5-arg builtin
#endif

// Async-DMA a [16 x C2] f32 tile (contiguous rows at gsrc) into LDS rows of
// stride ASTR floats, via one iterate-mode TDM descriptor (16 iterations of a
// 1 x C2 tile; lds_addr += ASTR, global_addr += C2 per iteration).
template <int C2>
__device__ __forceinline__ void tdm_load_tile(float* myB, const float* gsrc)
{
#if HAVE_TDM
    const unsigned lds = (unsigned)(size_t)myB;  // flat addr low 32 = LDS byte addr
    const unsigned long long ga = (unsigned long long)(size_t)gsrc;
    u32x4 g0;
    g0[0] = 1u;                                             // count=1 (valid)
    g0[1] = lds;                                            // lds_addr
    g0[2] = (unsigned)(ga & 0xFFFFFFFFull);                 // global_addr lo
    g0[3] = (unsigned)((ga >> 32) & 0x01FFFFFFull)          // global_addr hi
            | (2u << 30);                                   // type = 2 (image)
    i32x8 g1;
    g1[0] = (2 << 16) | (1 << 19);       // data_size=4B, iterate_enable
    g1[1] = (C2 & 0xFFFF) << 16;         // tensor_dim0[15:0] = C2
    g1[2] = (1 << 16);                   // tensor_dim1 = 1
    g1[3] = (C2 << 16);                  // tile_dim0 = C2
    g1[4] = 1;                           // tile_dim1 = 1, tile_dim2 = 0
    g1[5] = C2;                          // tensor_dim0_stride = C2
    g1[6] = 0;
    g1[7] = 0;
    i32x4 g2;
    g2[0] = 0;                           // tensor_dim2
    g2[1] = ASTR;                        // lds_addr_increment (data_size units)
    g2[2] = C2;                          // global_addr_increment (units)
    g2[3] = (15 << 16);                  // iterate_count = 15 -> 16 iterations
    i32x4 g3 = {0, 0, 0, 0};
#if TDM_ARITY6
    i32x8 g4 = {0, 0, 0, 0, 0, 0, 0, 0};
    __builtin_amdgcn_tensor_load_to_lds(g0, g1, g2, g3, g4, 0);
#else
    __builtin_amdgcn_tensor_load_to_lds(g0, g1, g2, g3, 0);
#endif
#endif
}

// 16x16 tile GEMM on the WMMA pipe. A from per-wave LDS tile (zero-padded to
// KP), B from block-shared pair-interleaved padded weights, fp32 accumulate.
// A 16x4 f32 frag: lanes0-15 {K=k,k+1}, lanes16-31 {K=k+2,k+3}
// B 4x16 f32 frag: V0 = rows {k,k+2}, V1 = rows {k+1,k+3} (row-striped halves)
template <int NCOLS, int KP, int WS>
__device__ __forceinline__ v8f tile_gemm(const float* __restrict__ aT,
                                         const float* __restrict__ sW,
                                         const float* __restrict__ bias,
                                         int n0, int lane)
{
    const bool hi  = lane >= 16;
    const int rowA = lane & 15;
    const int ncol = n0 + (lane & 15);
    const float bval = (ncol < NCOLS) ? bias[ncol] : 0.0f;
    v8f acc;
#pragma unroll
    for (int j = 0; j < 8; ++j) acc[j] = bval;

#pragma unroll
    for (int k = 0; k < KP; k += 4) {
        const int ka = k + (hi ? 2 : 0);
        const v2f a = *(const v2f*)&aT[rowA * ASTR + ka];
        const v2f b = *(const v2f*)&sW[(ka >> 1) * WS + 2 * ncol];
        acc = __builtin_amdgcn_wmma_f32_16x16x4_f32(
            false, a, false, b, (short)0, acc, false, false);
    }
    return acc;
}

// Write C fragment with ReLU into per-wave packed output tile [16][NOUT].
template <int NCOLS, int NOUT>
__device__ __forceinline__ void stage_tile(float* __restrict__ myO, v8f acc,
                                           int cbase, int n0, int lane)
{
    const int nl = lane & 15;
    const int ncol = n0 + nl;
    const int mBase = (lane >= 16) ? 8 : 0;
    if (ncol < NCOLS) {   // single mask for all 8 stores; vacuous when n0 + 16 <= NCOLS
#pragma unroll
        for (int j = 0; j < 8; ++j) {
            const float v = acc[j];
            myO[(mBase + j) * NOUT + cbase + ncol] = v > 0.0f ? v : 0.0f;
        }
    }
}

// One wave = one 16-row tile. EDGE: A1 row r = h[src[r]] + h[dst[r]];
// node: A1 row r = A1[r]. A2 is a plain [rows,C2] matrix (ea_in / agg).
template <int C1, int N1, int C2, int N2, bool EDGE>
__global__ __launch_bounds__(256) void fused_layer_kernel(
    const float* __restrict__ A1, const int* __restrict__ srcIdx,
    const int* __restrict__ dstIdx, const float* __restrict__ A2,
    const float* __restrict__ W1, const float* __restrict__ b1,
    const float* __restrict__ W2, const float* __restrict__ b2,
    float* __restrict__ out, int rows)
{
    constexpr int KP1 = (C1 + 3) & ~3;          // K padded to x4
    constexpr int KP2 = (C2 + 3) & ~3;
    constexpr int NP1 = (N1 + 15) & ~15;        // N padded to x16
    constexpr int NP2 = (N2 + 15) & ~15;
    constexpr int W1S = 2 * NP1 + 4;            // pair-row stride (floats)
    constexpr int W2S = 2 * NP2 + 4;
    constexpr int NOUT = N1 + N2;
    constexpr bool TDM_OK = (HAVE_TDM != 0) && ((C2 & 3) == 0);  // C2 == KP2

    __shared__ float sW1[(KP1 / 2) * W1S];
    __shared__ float sW2[(KP2 / 2) * W2S];
    __shared__ float sA[WAVES_PER_BLOCK * TILE_M * ASTR];   // GEMM1 operand
    __shared__ float sB[WAVES_PER_BLOCK * TILE_M * ASTR];   // GEMM2 operand
    __shared__ float sO[WAVES_PER_BLOCK * TILE_M * NOUT];   // output staging

    const int tid  = threadIdx.x;
    const int lane = tid & 31;
    const int wave = tid >> 5;
    float* myT = &sA[wave * TILE_M * ASTR];
    float* myB = &sB[wave * TILE_M * ASTR];
    float* myO = &sO[wave * TILE_M * NOUT];
    const int r0 = (blockIdx.x * WAVES_PER_BLOCK + wave) * TILE_M;  // < 2^21
    const bool fullTile = (r0 + TILE_M <= rows);

    // ---- Kick off async TDM of the GEMM2 operand tile (overlaps GEMM1) -----
    if (TDM_OK && fullTile)
        tdm_load_tile<C2>(myB, A2 + (size_t)r0 * C2);

    // ---- Stage zero-padded weights in LDS (pair-interleaved) ---------------
    for (int idx = tid; idx < (KP1 / 2) * NP1; idx += 256) {
        const int kp = idx / NP1, n = idx - kp * NP1, k0 = 2 * kp;
        sW1[kp * W1S + 2 * n + 0] = (k0 < C1 && n < N1) ? W1[k0 * N1 + n] : 0.0f;
        sW1[kp * W1S + 2 * n + 1] = (k0 + 1 < C1 && n < N1) ? W1[(k0 + 1) * N1 + n] : 0.0f;
    }
    for (int idx = tid; idx < (KP2 / 2) * NP2; idx += 256) {
        const int kp = idx / NP2, n = idx - kp * NP2, k0 = 2 * kp;
        sW2[kp * W2S + 2 * n + 0] = (k0 < C2 && n < N2) ? W2[k0 * N2 + n] : 0.0f;
        sW2[kp * W2S + 2 * n + 1] = (k0 + 1 < C2 && n < N2) ? W2[(k0 + 1) * N2 + n] : 0.0f;
    }

    // ---- Phase A: stage A1 tile (gather-add for edge layers) ---------------
    const int r = lane >> 1;       // 2 lanes per row
    const int cp = lane & 1;       // column parity
    const int row = r0 + r;
    if (row < rows) {
        if (EDGE) {
            const int s = srcIdx[row], d = dstIdx[row];
            const float* ps = A1 + (size_t)s * C1;
            const float* pd = A1 + (size_t)d * C1;
            if constexpr ((C1 & 3) == 0) {      // vectorized gather
#pragma unroll
                for (int c = 4 * cp; c < C1; c += 8) {
                    const float4 a = *(const float4*)&ps[c];
                    const float4 b = *(const float4*)&pd[c];
                    float4 s4;
                    s4.x = a.x + b.x; s4.y = a.y + b.y;
                    s4.z = a.z + b.z; s4.w = a.w + b.w;
                    *(float4*)&myT[r * ASTR + c] = s4;
                }
            } else {
#pragma unroll
                for (int c = cp; c < C1; c += 2) myT[r * ASTR + c] = ps[c] + pd[c];
            }
        } else {
            const float* p = A1 + (size_t)row * C1;
            if constexpr ((C1 & 3) == 0) {
#pragma unroll
                for (int c = 4 * cp; c < C1; c += 8)
                    *(float4*)&myT[r * ASTR + c] = *(const float4*)&p[c];
            } else {
#pragma unroll
                for (int c = cp; c < C1; c += 2) myT[r * ASTR + c] = p[c];
            }
        }
    } else {
#pragma unroll
        for (int c = cp; c < C1; c += 2) myT[r * ASTR + c] = 0.0f;
    }
#pragma unroll
    for (int c = C1 + cp; c < KP1; c += 2) myT[r * ASTR + c] = 0.0f;
    __syncthreads();

    // ---- GEMM 1: [16,C1] x [C1,N1] -> tile cols [0,N1) ---------------------
#pragma unroll
    for (int nt = 0; nt < NP1 / 16; ++nt) {
        const v8f acc = tile_gemm<N1, KP1, W1S>(myT, sW1, b1, nt * 16, lane);
        stage_tile<N1, NOUT>(myO, acc, 0, nt * 16, lane);
    }

    // ---- Phase B: GEMM2 operand -> myB (manual path when TDM not used) -----
    if (!(TDM_OK && fullTile)) {
        if (row < rows) {
            const float* p = A2 + (size_t)row * C2;
            if constexpr ((C2 & 3) == 0) {
#pragma unroll
                for (int c = 4 * cp; c < C2; c += 8)
                    *(float4*)&myB[r * ASTR + c] = *(const float4*)&p[c];
            } else {
#pragma unroll
                for (int c = cp; c < C2; c += 2) myB[r * ASTR + c] = p[c];
            }
        } else {
#pragma unroll
            for (int c = cp; c < C2; c += 2) myB[r * ASTR + c] = 0.0f;
        }
#pragma unroll
        for (int c = C2 + cp; c < KP2; c += 2) myB[r * ASTR + c] = 0.0f;
    }
#if HAVE_TDM
    if (TDM_OK && fullTile) __builtin_amdgcn_s_wait_tensorcnt(0);
#endif
    __syncthreads();

    // ---- GEMM 2: [16,C2] x [C2,N2] -> tile cols [N1,NOUT) ------------------
#pragma unroll
    for (int nt = 0; nt < NP2 / 16; ++nt) {
        const v8f acc = tile_gemm<N2, KP2, W2S>(myB, sW2, b2, nt * 16, lane);
        stage_tile<N2, NOUT>(myO, acc, N1, nt * 16, lane);
    }
    __syncthreads();

    // ---- Flush: the 16-row tile is a contiguous span of 16*NOUT floats -----
    if (fullTile) {
        float4* __restrict__ dst4 = (float4*)(out + (size_t)r0 * NOUT);
        const float4* __restrict__ src4 = (const float4*)myO;
        constexpr int NV4 = TILE_M * NOUT / 4;       // always integral
#pragma unroll
        for (int i = lane; i < NV4; i += 32) dst4[i] = src4[i];
    } else {                                         // ragged last tile
        for (int i = lane; i < TILE_M * NOUT; i += 32) {
            const int rr = i / NOUT;
            if (r0 + rr < rows) out[(size_t)r0 * NOUT + i] = myO[i];
        }
    }
}

__global__ void fill_zero_kernel(float* __restrict__ p, size_t n)
{
    size_t i = (size_t)blockIdx.x * blockDim.x + threadIdx.x;
    if (i < n) p[i] = 0.0f;
}

// agg[src[e], c] += ea[e, c]; CE compile-time so /CE is a mul-shift.
template <int CE>
__global__ void seg_sum_kernel(const float* __restrict__ ea,
                               const int* __restrict__ src,
                               float* __restrict__ agg, int E_)
{
    const int t = blockIdx.x * blockDim.x + threadIdx.x;
    const int e = t / CE;
    if (e < E_) {
        const int c = t - e * CE;
        atomicAdd(&agg[(size_t)src[e] * CE + c], ea[(size_t)e * CE + c]);
    }
}

__global__ void state_reduce_kernel(const float* __restrict__ h,
                                    float* __restrict__ state, int n)
{
    __shared__ float part[6];
    if (threadIdx.x < 6) part[threadIdx.x] = 0.0f;
    __syncthreads();
    const int i = blockIdx.x * blockDim.x + threadIdx.x;
    if (i < n) {
#pragma unroll
        for (int j = 0; j < 6; ++j) atomicAdd(&part[j], h[(size_t)i * 6 + j]);
    }
    __syncthreads();
    if (threadIdx.x < 6) atomicAdd(&state[threadIdx.x], part[threadIdx.x]);
}

__global__ void final_concat_kernel(const float* __restrict__ h,
                                    const float* __restrict__ state,
                                    float* __restrict__ out, int n)
{
    const int i = blockIdx.x * blockDim.x + threadIdx.x;
    if (i < n) {
#pragma unroll
        for (int j = 0; j < 6; ++j) {
            out[(size_t)i * 12 + j]     = state[j];
            out[(size_t)i * 12 + 6 + j] = h[(size_t)i * 6 + j];
        }
    }
}

extern "C" void kernel_launch(void* const* d_in, const int* in_sizes, int n_in,
                              void* d_out, int out_size, void* d_ws, size_t ws_size,
                              hipStream_t stream)
{
    const float* x     = (const float*)d_in[0];
    const int*   eidx  = (const int*)d_in[1];
    const float* eattr = (const float*)d_in[2];
    const int N_ = in_sizes[0] / 5;
    const int E_ = in_sizes[1] / 2;
    const int* src = eidx;
    const int* dst = eidx + E_;

    // params flattened as pytree: per layer, dict keys sorted: We, Wx, be, bx
    const float *We[10], *Wx[10], *be[10], *bx[10];
    for (int i = 0; i < 10; ++i) {
        We[i] = (const float*)d_in[3 + 4 * i + 0];
        Wx[i] = (const float*)d_in[3 + 4 * i + 1];
        be[i] = (const float*)d_in[3 + 4 * i + 2];
        bx[i] = (const float*)d_in[3 + 4 * i + 3];
    }

    // workspace carving (floats)
    float* ws = (float*)d_ws;
    size_t off = 0;
    float* eaA = ws + off; off += (size_t)E_ * 48;
    float* eaB = ws + off; off += (size_t)E_ * 48;
    float* hA  = ws + off; off += (size_t)N_ * 56;
    float* hB  = ws + off; off += (size_t)N_ * 56;
    float* agg = ws + off; off += (size_t)N_ * 48;
    float* st  = ws + off; off += 64;

    const dim3 blk(256);
    const unsigned gridE = (E_ + ROWS_PER_BLOCK - 1) / ROWS_PER_BLOCK;
    const unsigned gridN = (N_ + ROWS_PER_BLOCK - 1) / ROWS_PER_BLOCK;

#define EDGE_LAYER(i, C1, N1, C2, N2, HIN, EAIN, EAOUT)                        \
    fused_layer_kernel<C1, N1, C2, N2, true><<<gridE, blk, 0, stream>>>(       \
        HIN, src, dst, EAIN, Wx[i], bx[i], We[i], be[i], EAOUT, E_)

#define NODE_LAYER(i, C1, N1, C2, N2, HIN, EAIN, HOUT)                         \
    do {                                                                       \
        const size_t aggN = (size_t)N_ * C2;                                   \
        fill_zero_kernel<<<(unsigned)((aggN + 255) / 256), blk, 0, stream>>>(  \
            agg, aggN);                                                        \
        const size_t tot = (size_t)E_ * C2;                                    \
        seg_sum_kernel<C2><<<(unsigned)((tot + 255) / 256), blk, 0, stream>>>( \
            EAIN, src, agg, E_);                                               \
        fused_layer_kernel<C1, N1, C2, N2, false><<<gridN, blk, 0, stream>>>(  \
            HIN, nullptr, nullptr, agg, Wx[i], bx[i], We[i], be[i], HOUT, N_); \
    } while (0)

    EDGE_LAYER(0, 5, 6, 1, 2, x, eattr, eaA);          // ea: [E,8]
    NODE_LAYER(1, 5, 6, 8, 10, x, eaA, hA);            // h:  [N,16]
    EDGE_LAYER(2, 16, 32, 8, 16, hA, eaA, eaB);        // ea: [E,48]
    NODE_LAYER(3, 16, 32, 48, 24, hA, eaB, hB);        // h:  [N,56]
    EDGE_LAYER(4, 56, 24, 48, 24, hB, eaB, eaA);       // ea: [E,48]
    NODE_LAYER(5, 56, 24, 48, 24, hB, eaA, hA);        // h:  [N,48]
    EDGE_LAYER(6, 48, 13, 48, 13, hA, eaA, eaB);       // ea: [E,26]
    NODE_LAYER(7, 48, 13, 26, 8, hA, eaB, hB);         // h:  [N,21]
    EDGE_LAYER(8, 21, 3, 26, 3, hB, eaB, eaA);         // ea: [E,6]
    NODE_LAYER(9, 21, 3, 6, 3, hB, eaA, hA);           // h:  [N,6]

#undef EDGE_LAYER
#undef NODE_LAYER

    fill_zero_kernel<<<1, 64, 0, stream>>>(st, 64);
    state_reduce_kernel<<<(N_ + 255) / 256, blk, 0, stream>>>(hA, st, N_);
    final_concat_kernel<<<(N_ + 255) / 256, blk, 0, stream>>>(hA, st, (float*)d_out, N_);
}